// Patchify_15040975471281
// MI455X (gfx1250) — compile-verified
//
#include <hip/hip_runtime.h>

typedef __attribute__((ext_vector_type(2))) float v2f;
typedef __attribute__((ext_vector_type(8))) float v8f;

#define K_TOT 96
#define LDS_STRIDE 100   // pad 96 -> 100: conflict-free b64 LDS reads

__global__ __launch_bounds__(128) void patchify_gemm_f32wmma(
    const float* __restrict__ x, const float* __restrict__ Wm,
    const float* __restrict__ bias, float* __restrict__ out)
{
    __shared__ __align__(16) float ldsA[16 * LDS_STRIDE];

    const int n0 = blockIdx.x * 16;   // 196 tiles of 16 positions (3136 = 196*16)
    const int b  = blockIdx.y;        // batch 0..63
    const float* xb = x + (size_t)b * (3 * 224 * 224);

    // ---- Phase 1: gather A tile (16 rows x 96 K) from x into LDS ----
    // cat channel k: k<48 -> unfold(pad=0,dil=1); k>=48 -> unfold(pad=2,dil=2)
    // within each half: c = k/16, i = (k/4)%4, j = k%4  (from transpose(0,1,3,5,2,4))
    for (int e = threadIdx.x; e < 16 * K_TOT; e += 128) {
        int r  = e / K_TOT;
        int k  = e - r * K_TOT;
        int n  = n0 + r;
        int ho = n / 56;
        int wo = n - ho * 56;
        int kk = k & 47;
        int c = kk >> 4, i = (kk >> 2) & 3, j = kk & 3;
        float val;
        if (k < 48) {
            int h = ho * 4 + i, w = wo * 4 + j;       // always in [0,224)
            val = xb[(c * 224 + h) * 224 + w];
        } else {
            int h = ho * 4 + 2 * i - 2, w = wo * 4 + 2 * j - 2;
            bool ok = (h >= 0) & (h < 224) & (w >= 0) & (w < 224);
            val = ok ? xb[(c * 224 + h) * 224 + w] : 0.0f;
        }
        ldsA[r * LDS_STRIDE + k] = val;
    }
    __syncthreads();

    const int lane = threadIdx.x & 31;
    const int wave = threadIdx.x >> 5;    // 0..3, each owns 192 output cols
    const int row  = lane & 15;           // M (for A) / N (for B,C) lane index
    const int koff = (lane >> 4) << 1;    // upper half-wave holds K+2,K+3

    // ---- load A fragments once; reuse across all 12 N-subtiles ----
    // A 16x4 f32 layout: lane<16 {K=0,K=1}, lane>=16 {K=2,K=3}, M = lane%16
    v2f a[24];
    #pragma unroll
    for (int t = 0; t < 24; ++t)
        a[t] = *(const v2f*)&ldsA[row * LDS_STRIDE + t * 4 + koff];

    // ---- Phase 2: 12 subtiles of 16 columns per wave ----
    for (int s = 0; s < 12; ++s) {
        const int d0 = wave * 192 + s * 16;
        v8f acc = {};
        // B 4x16 layout mirrors A: lane pair holds B[kbase+koff][d0+row], B[kbase+koff+1][d0+row]
        const float* wp = Wm + (size_t)(d0 + row) * K_TOT + koff;
        #pragma unroll
        for (int t = 0; t < 24; ++t) {
            v2f bf = *(const v2f*)(wp + t * 4);
            acc = __builtin_amdgcn_wmma_f32_16x16x4_f32(
                false, a[t], false, bf, (short)0, acc, false, false);
        }
        const float bv = bias[d0 + row];   // bias depends only on N = row
        // C/D layout: VGPR v -> M = v + (lane>=16 ? 8 : 0), N = row
        const int mhi = (lane >> 4) << 3;
        const size_t base = ((size_t)b * 3136 + n0) * 768 + (size_t)(d0 + row);
        #pragma unroll
        for (int v = 0; v < 8; ++v) {
            // non-temporal: keep the 616MB output stream from evicting L2-resident x
            __builtin_nontemporal_store(acc[v] + bv,
                                        out + base + (size_t)(v + mhi) * 768);
        }
    }
}

extern "C" void kernel_launch(void* const* d_in, const int* in_sizes, int n_in,
                              void* d_out, int out_size, void* d_ws, size_t ws_size,
                              hipStream_t stream) {
    const float* x    = (const float*)d_in[0];   // (64,3,224,224) f32
    const float* Wm   = (const float*)d_in[1];   // (768,96) f32
    const float* bias = (const float*)d_in[2];   // (768,) f32
    float* out = (float*)d_out;                  // (64,3136,768) f32

    dim3 grid(196, 64);   // 196 position-tiles x 64 batches
    patchify_gemm_f32wmma<<<grid, 128, 0, stream>>>(x, Wm, bias, out);
}